// LearnableFF_44212393345585
// MI455X (gfx1250) — compile-verified
//
#include <hip/hip_runtime.h>
#include <hip/hip_bf16.h>

#define BB 64
#define LL 128
#define RR 2048
#define FF 128
#define N_STEPS 50
#define LRATE 0.01f
#define BETA1 0.9f
#define BETA2 0.999f
#define EPSA 1e-8f
#define WDEC 0.01f

#define NTHREADS 1024
#define NWAVES (NTHREADS / 32)

typedef __attribute__((ext_vector_type(2))) float v2f;
typedef __attribute__((ext_vector_type(8))) float v8f;

// ---------------------------------------------------------------------------
// Kernel 1: atn[b] = lig_feat[b] (128x128) @ rec_feat[b]^T (128x2048), f32 WMMA.
// One wave per 16x16 output tile, K=128 via 32 chained V_WMMA_F32_16X16X4_F32.
// A 16x4 f32 layout: lanes 0-15 hold row M=lane, K={0,1}; lanes 16-31 K={2,3}.
// B 4x16 layout mirrors it with N=lane%16. C: VGPR i -> row i + 8*(lane/16).
// ---------------------------------------------------------------------------
__global__ __launch_bounds__(256) void atn_wmma_kernel(
    const float* __restrict__ lf, const float* __restrict__ rf,
    float* __restrict__ atn)
{
    const int b    = blockIdx.z;
    const int li   = blockIdx.y;                 // 0..7  (L tile)
    const int wv   = threadIdx.x >> 5;           // 0..7
    const int rj   = blockIdx.x * 8 + wv;        // 0..127 (R tile)
    const int lane = threadIdx.x & 31;
    const int half = lane >> 4;
    const int m    = lane & 15;

    const float* Arow = lf + ((size_t)b * LL + li * 16 + m) * FF;
    const float* Brow = rf + ((size_t)b * RR + rj * 16 + m) * FF;

    v8f c = {0.f, 0.f, 0.f, 0.f, 0.f, 0.f, 0.f, 0.f};
#pragma unroll
    for (int k0 = 0; k0 < FF; k0 += 4) {
        v2f a  = *(const v2f*)(Arow + k0 + 2 * half);
        v2f bv = *(const v2f*)(Brow + k0 + 2 * half);
        c = __builtin_amdgcn_wmma_f32_16x16x4_f32(
                false, a, false, bv, (short)0, c, false, false);
    }

    float* Cp = atn + (((size_t)b * LL + li * 16) * RR) + (size_t)rj * 16 + m;
#pragma unroll
    for (int i = 0; i < 8; ++i)
        Cp[(size_t)(i + 8 * half) * RR] = c[i];
}

// ---------------------------------------------------------------------------
// Kernel 2: per-batch 50-step AdamW on (rot, trans). One block per batch.
// 1024 threads = 32 waves (8 waves/SIMD for latency hiding); wave w handles
// rows l = w, w+32, ... (4 rows), lanes stride over r (coalesced 128B atn
// reads from L2, single ds_load_b128 per r for packed centered rc).
// ---------------------------------------------------------------------------
__global__ __launch_bounds__(NTHREADS) void opt_kernel(
    const float* __restrict__ lig_coord, const float* __restrict__ rec_coord,
    const float* __restrict__ rot_init,  const float* __restrict__ trans_init,
    const float* __restrict__ atn, float* __restrict__ out)
{
    __shared__ float4 rc4[RR];                 // centered rec coords (xyz, pad)
    __shared__ float l0x[LL], l0y[LL], l0z[LL];
    __shared__ float lcx[LL], lcy[LL], lcz[LL];
    __shared__ float P[12];                    // rot[0..8], tr[9..11]
    __shared__ float Mm[12], Vv[12];           // Adam moments
    __shared__ float red[NWAVES][14];          // per-wave: gr(9) gt(3) U(1)
    __shared__ float meanr[3], meanl[3];
    __shared__ float Uout;

    const int b    = blockIdx.x;
    const int tid  = threadIdx.x;
    const int wave = tid >> 5;
    const int lane = tid & 31;

    // ---- load coords to LDS ----
    for (int r = tid; r < RR; r += NTHREADS) {
        const float* p = rec_coord + ((size_t)b * RR + r) * 3;
        rc4[r] = make_float4(p[0], p[1], p[2], 0.f);
    }
    if (tid < LL) {
        const float* p = lig_coord + ((size_t)b * LL + tid) * 3;
        l0x[tid] = p[0]; l0y[tid] = p[1]; l0z[tid] = p[2];
    }
    if (tid < 9)  { P[tid] = rot_init[(size_t)b * 9 + tid]; }
    if (tid < 3)  { P[9 + tid] = trans_init[(size_t)b * 3 + tid]; }
    if (tid < 12) { Mm[tid] = 0.f; Vv[tid] = 0.f; }
    __syncthreads();

    // ---- centering (one-time; thread 0 serial is negligible) ----
    if (tid == 0) {
        float sx = 0.f, sy = 0.f, sz = 0.f;
        for (int r = 0; r < RR; ++r) { sx += rc4[r].x; sy += rc4[r].y; sz += rc4[r].z; }
        meanr[0] = sx / RR; meanr[1] = sy / RR; meanr[2] = sz / RR;
        sx = sy = sz = 0.f;
        for (int l = 0; l < LL; ++l) { sx += l0x[l]; sy += l0y[l]; sz += l0z[l]; }
        meanl[0] = sx / LL; meanl[1] = sy / LL; meanl[2] = sz / LL;
    }
    __syncthreads();
    for (int r = tid; r < RR; r += NTHREADS) {
        float4 c = rc4[r];
        rc4[r] = make_float4(c.x - meanr[0], c.y - meanr[1], c.z - meanr[2], 0.f);
    }
    if (tid < LL) { l0x[tid] -= meanl[0]; l0y[tid] -= meanl[1]; l0z[tid] -= meanl[2]; }

    const float inv_lr2 = 1.0f / ((float)LL * (float)RR);
    float b1t = 1.f, b2t = 1.f;

    for (int t = 1; t <= N_STEPS; ++t) {
        b1t *= BETA1; b2t *= BETA2;
        __syncthreads();
        // lc = rot @ lc0c + tr
        if (tid < LL) {
            float x = l0x[tid], y = l0y[tid], z = l0z[tid];
            lcx[tid] = fmaf(P[0], x, fmaf(P[1], y, fmaf(P[2], z, P[9])));
            lcy[tid] = fmaf(P[3], x, fmaf(P[4], y, fmaf(P[5], z, P[10])));
            lcz[tid] = fmaf(P[6], x, fmaf(P[7], y, fmaf(P[8], z, P[11])));
        }
        __syncthreads();

        float gr[9] = {0,0,0,0,0,0,0,0,0};
        float gt[3] = {0,0,0};
        float Uacc  = 0.f;

        for (int l = wave; l < LL; l += NWAVES) {
            const float plx = lcx[l], ply = lcy[l], plz = lcz[l];
            const float* arow = atn + ((size_t)b * LL + l) * RR;
            // glc[l] = sum_r coef * (lc[l] - rc[r]) accumulated directly
            float Gx = 0.f, Gy = 0.f, Gz = 0.f;
#pragma unroll 4
            for (int r = lane; r < RR; r += 32) {
                float4 c  = rc4[r];                       // ds_load_b128
                float dx = plx - c.x, dy = ply - c.y, dz = plz - c.z;
                float sq = fmaf(dx, dx, fmaf(dy, dy, dz * dz));
                float rs = __builtin_amdgcn_rcpf(sq);
                float w  = arow[r] * rs;                  // coalesced b32
                float cf = w * rs;
                Uacc += w;
                Gx = fmaf(cf, dx, Gx);
                Gy = fmaf(cf, dy, Gy);
                Gz = fmaf(cf, dz, Gz);
            }
#pragma unroll
            for (int off = 16; off > 0; off >>= 1) {
                Gx += __shfl_xor(Gx, off, 32);
                Gy += __shfl_xor(Gy, off, 32);
                Gz += __shfl_xor(Gz, off, 32);
            }
            // gr += glc (x) lc0c,  gt += glc    (scale -2/(L*R) applied later)
            float x0 = l0x[l], y0 = l0y[l], z0 = l0z[l];
            gr[0] = fmaf(Gx, x0, gr[0]); gr[1] = fmaf(Gx, y0, gr[1]); gr[2] = fmaf(Gx, z0, gr[2]);
            gr[3] = fmaf(Gy, x0, gr[3]); gr[4] = fmaf(Gy, y0, gr[4]); gr[5] = fmaf(Gy, z0, gr[5]);
            gr[6] = fmaf(Gz, x0, gr[6]); gr[7] = fmaf(Gz, y0, gr[7]); gr[8] = fmaf(Gz, z0, gr[8]);
            gt[0] += Gx; gt[1] += Gy; gt[2] += Gz;
        }
#pragma unroll
        for (int off = 16; off > 0; off >>= 1) Uacc += __shfl_xor(Uacc, off, 32);
        if (lane == 0) {
#pragma unroll
            for (int i = 0; i < 9; ++i) red[wave][i] = gr[i];
            red[wave][9]  = gt[0]; red[wave][10] = gt[1]; red[wave][11] = gt[2];
            red[wave][12] = Uacc;
        }
        __syncthreads();
        if (tid == 0) {
            float g[12]; float U = 0.f;
#pragma unroll
            for (int i = 0; i < 12; ++i) g[i] = 0.f;
            for (int w = 0; w < NWAVES; ++w) {
#pragma unroll
                for (int i = 0; i < 12; ++i) g[i] += red[w][i];
                U += red[w][12];
            }
            U *= inv_lr2;
            if (t == N_STEPS) Uout = U;     // U recorded BEFORE this step's update
            const float gscale = -2.0f * inv_lr2;
#pragma unroll
            for (int i = 0; i < 12; ++i) {
                float gi = g[i] * gscale;
                float p  = P[i] * (1.0f - LRATE * WDEC);
                float m  = BETA1 * Mm[i] + (1.0f - BETA1) * gi;
                float v  = BETA2 * Vv[i] + (1.0f - BETA2) * gi * gi;
                Mm[i] = m; Vv[i] = v;
                float mh = m / (1.0f - b1t);
                float vh = v / (1.0f - b2t);
                P[i] = p - LRATE * mh / (sqrtf(vh) + EPSA);
            }
        }
        __syncthreads();
    }
    if (tid == 0) out[b] = -Uout;
}

extern "C" void kernel_launch(void* const* d_in, const int* in_sizes, int n_in,
                              void* d_out, int out_size, void* d_ws, size_t ws_size,
                              hipStream_t stream) {
    (void)in_sizes; (void)n_in; (void)out_size; (void)ws_size;
    const float* lf   = (const float*)d_in[0];   // [B,L,F]
    const float* rf   = (const float*)d_in[1];   // [B,R,F]
    const float* lc0  = (const float*)d_in[2];   // [B,L,3]
    const float* rc   = (const float*)d_in[3];   // [B,R,3]
    const float* rot0 = (const float*)d_in[4];   // [B,3,3]
    const float* tr0  = (const float*)d_in[5];   // [B,3]
    float* out = (float*)d_out;                  // [B]
    float* atn = (float*)d_ws;                   // [B,L,R] f32 = 64 MB

    dim3 g1(RR / 128, LL / 16, BB);              // (16, 8, 64), wave-per-tile
    atn_wmma_kernel<<<g1, 256, 0, stream>>>(lf, rf, atn);
    opt_kernel<<<BB, NTHREADS, 0, stream>>>(lc0, rc, rot0, tr0, atn, out);
}